// StructureAwareGCN_40647570489409
// MI455X (gfx1250) — compile-verified
//
#include <hip/hip_runtime.h>

typedef float v2f __attribute__((ext_vector_type(2)));
typedef float v8f __attribute__((ext_vector_type(8)));
typedef unsigned int u32x4 __attribute__((ext_vector_type(4)));
typedef int i32x4 __attribute__((ext_vector_type(4)));
typedef int i32x8 __attribute__((ext_vector_type(8)));

constexpr int N = 8192;   // nodes
constexpr int D = 512;    // feature dim
constexpr int B = 4096;   // batch (output rows)

#if __has_builtin(__builtin_amdgcn_tensor_load_to_lds) && \
    __has_builtin(__builtin_amdgcn_s_wait_tensorcnt)
#define USE_TDM 1
#else
#define USE_TDM 0
#endif

static __device__ __forceinline__ v8f wmma4(v2f a, v2f b, v8f c) {
  // V_WMMA_F32_16X16X4_F32 : D(16x16,f32) = A(16x4,f32) * B(4x16,f32) + C
  return __builtin_amdgcn_wmma_f32_16x16x4_f32(
      false, a, false, b, (short)0, c, false, false);
}

#if USE_TDM
// Issue a TDM load of a 16-row x 64-DWORD fp32 tile of A into LDS, with LDS
// padding of 4 DWORDs after every 64 DWORDs (row stride 68 floats -> no bank
// conflicts for the WMMA A-fragment ds reads). D# layout per CDNA5 ISA ch.8.
static __device__ __forceinline__ void tdm_load_tile(const float* gsrc,
                                                     unsigned lds_byte_addr) {
  unsigned long long ga = (unsigned long long)(size_t)gsrc;
  u32x4 g0;
  g0[0] = 1u;                                            // count=1 (valid user D#)
  g0[1] = lds_byte_addr;                                 // lds_addr
  g0[2] = (unsigned)(ga & 0xffffffffull);                // global_addr[31:0]
  g0[3] = (unsigned)((ga >> 32) & 0x01ffffffull)         // global_addr[56:32]
          | 0x80000000u;                                 // type=2 ("image")
  i32x8 g1;
  g1[0] = (2 << 16)      // data_size = 4 bytes
        | (1 << 20)      // pad_enable
        | (5 << 22)      // pad_interval: 64 DWORDs
        | (3 << 25);     // pad_amount : 4 DWORDs
  g1[1] = (N & 0xffff) << 16;                            // tensor_dim0[15:0]
  g1[2] = ((N >> 16) & 0xffff) | ((N & 0xffff) << 16);   // dim0[31:16] | dim1[15:0]
  g1[3] = ((N >> 16) & 0xffff) | (64 << 16);             // dim1[31:16] | tile_dim0=64
  g1[4] = 16;                                            // tile_dim1=16, tile_dim2=0
  g1[5] = N;                                             // tensor_dim0_stride[31:0]
  g1[6] = 0;                                             // stride0[47:32] | dim1_stride lo
  g1[7] = 0;
  i32x4 z4 = {0, 0, 0, 0};
#if __clang_major__ >= 23
  i32x8 z8 = {0, 0, 0, 0, 0, 0, 0, 0};
  __builtin_amdgcn_tensor_load_to_lds(g0, g1, z4, z4, z8, 0);
#else
  __builtin_amdgcn_tensor_load_to_lds(g0, g1, z4, z4, 0);
#endif
}
#endif  // USE_TDM

// ---------------------------------------------------------------------------
// Kernel 1: dinv[i] = rsqrt(rowsum(A[i,:]) + 1e-8). One block per row.
// ---------------------------------------------------------------------------
__global__ void __launch_bounds__(256) k_degree(const float* __restrict__ A,
                                                float* __restrict__ dinv) {
  const int row = blockIdx.x;
  const int tid = threadIdx.x;
  const float* rp = A + (size_t)row * N;
  float s = 0.f;
#pragma unroll
  for (int c = tid * 4; c < N; c += 256 * 4) {
    float4 v = *(const float4*)(rp + c);
    s += v.x + v.y + v.z + v.w;
  }
  for (int off = 16; off > 0; off >>= 1) s += __shfl_down(s, off, 32);
  __shared__ float red[8];
  if ((tid & 31) == 0) red[tid >> 5] = s;
  __syncthreads();
  if (tid == 0) {
    float t = 0.f;
#pragma unroll
    for (int w = 0; w < 8; ++w) t += red[w];
    dinv[row] = rsqrtf(t + 1e-8f);
  }
}

// ---------------------------------------------------------------------------
// Kernel 2: Hs[j,:] = dinv[j] * (X @ Wg^T)[j,:]    (all rows)
//           S[j,:]  =            (X @ Ws^T)[j,:]   (rows < B; DO_SKIP variant)
// One 16x16 tile per wave; software-pipelined fp32 WMMA over K=512.
// ---------------------------------------------------------------------------
template <bool DO_SKIP>
__global__ void __launch_bounds__(256) k_features(const float* __restrict__ X,
                                                  const float* __restrict__ Wg,
                                                  const float* __restrict__ Ws,
                                                  const float* __restrict__ dinv,
                                                  float* __restrict__ Hs,
                                                  float* __restrict__ S,
                                                  int tile_base) {
  const int lane  = threadIdx.x & 31;
  const int wid   = threadIdx.x >> 5;
  const int tile  = tile_base + blockIdx.x * 8 + wid;
  const int m0    = (tile >> 5) * 16;
  const int n0    = (tile & 31) * 16;
  const int nlane = lane & 15;
  const int basek = (lane >> 4) * 2;  // K-half select (ISA A/B fragment layout)

  const float* xrow  = X  + (size_t)(m0 + nlane) * D + basek;
  const float* wgrow = Wg + (size_t)(n0 + nlane) * D + basek;
  const float* wsrow = Ws + (size_t)(n0 + nlane) * D + basek;

  v8f cg = {};
  v8f cs = {};

  // 2-stage software pipeline: fetch k0+4 while multiplying k0.
  v2f a  = *(const v2f*)(xrow);
  v2f bg = *(const v2f*)(wgrow);
  v2f bs;
  if (DO_SKIP) bs = *(const v2f*)(wsrow);

#pragma unroll 8
  for (int k0 = 0; k0 < D; k0 += 4) {
    const int kn = (k0 + 4 < D) ? k0 + 4 : k0;  // last iter reloads (harmless)
    v2f an  = *(const v2f*)(xrow + kn);
    v2f bgn = *(const v2f*)(wgrow + kn);
    v2f bsn;
    if (DO_SKIP) bsn = *(const v2f*)(wsrow + kn);
    cg = wmma4(a, bg, cg);
    if (DO_SKIP) cs = wmma4(a, bs, cs);
    a = an; bg = bgn;
    if (DO_SKIP) bs = bsn;
  }

  const int rowoff = (lane >> 4) * 8;  // C/D layout: lanes>=16 hold M=r+8
#pragma unroll
  for (int r = 0; r < 8; ++r) {
    const int row = m0 + rowoff + r;
    const float dv = dinv[row];
    Hs[(size_t)row * D + n0 + nlane] = dv * cg[r];
    if (DO_SKIP) S[(size_t)row * D + n0 + nlane] = cs[r];
  }
}

// ---------------------------------------------------------------------------
// Kernel 3: G = A[:B,:] @ Hs (4096x8192 x 8192x512 fp32 WMMA), fused epilogue
//   out[i,d] = X[i,d] + 0.5 * relu(dinv[i]*G[i,d] + S[i,d])
// Block = 16 A-rows x 512 cols (8 waves x 4 N-tiles). A strip is DMA'd to LDS
// by the Tensor Data Mover (double-buffered, s_wait_tensorcnt) when available;
// Hs is streamed from L2. B-fragments are software-pipelined vs the WMMAs.
// ---------------------------------------------------------------------------
__global__ void __launch_bounds__(256) k_gcn(const float* __restrict__ A,
                                             const float* __restrict__ X,
                                             const float* __restrict__ Hs,
                                             const float* __restrict__ S,
                                             const float* __restrict__ dinv,
                                             float* __restrict__ out) {
  constexpr int KC = 64;
  constexpr int LROW = 68;            // padded LDS row stride (floats)
  __shared__ float lds_a[2][16 * LROW];

  const int tid   = threadIdx.x;
  const int lane  = tid & 31;
  const int wid   = tid >> 5;
  const int m0    = blockIdx.x * 16;
  const int nlane = lane & 15;
  const int basek = (lane >> 4) * 2;
  const int ncol0 = wid * 64;         // this wave's 4 N-tiles

  v8f acc[4] = {{}, {}, {}, {}};
  const int nchunks = N / KC;         // 128

#if USE_TDM
  const unsigned lbyte0 = (unsigned)(size_t)(void*)&lds_a[0][0];
  const unsigned lbyte1 = (unsigned)(size_t)(void*)&lds_a[1][0];
  const float* abase = A + (size_t)m0 * N;
  if (tid < 32) tdm_load_tile(abase, lbyte0);   // prime chunk 0 -> buf 0
#else
  const int ldrow = tid >> 4;          // 0..15 : strip row
  const int ldseg = tid & 15;          // 0..15 : 4-float segment
  const float* aload = A + (size_t)(m0 + ldrow) * N + ldseg * 4;
#endif

  for (int i = 0; i < nchunks; ++i) {
    const int kb = i * KC;
#if USE_TDM
    if (tid < 32) {
      if (i + 1 < nchunks) {
        // Previous end-of-iter barrier guarantees buf[(i+1)&1] is free.
        tdm_load_tile(abase + (size_t)(i + 1) * KC,
                      ((i + 1) & 1) ? lbyte1 : lbyte0);
        __builtin_amdgcn_s_wait_tensorcnt(1);   // chunk i complete (in-order)
      } else {
        __builtin_amdgcn_s_wait_tensorcnt(0);
      }
    }
    __syncthreads();
    const float* lbuf = &lds_a[i & 1][0];
#else
    __syncthreads();
    float4 av = *(const float4*)(aload + kb);
    *(float4*)(&lds_a[0][ldrow * LROW + ldseg * 4]) = av;
    if (i + 1 < nchunks) __builtin_prefetch(aload + kb + KC, 0, 0);
    __syncthreads();
    const float* lbuf = &lds_a[0][0];
#endif

    // Software-pipelined inner loop: prefetch next B-fragments into registers.
    v2f b[4], bn[4];
    {
      const float* hp0 = Hs + (size_t)(kb + basek) * D;
#pragma unroll
      for (int t = 0; t < 4; ++t) {
        const int col = ncol0 + t * 16 + nlane;
        b[t].x = hp0[col];
        b[t].y = hp0[D + col];
      }
    }
#pragma unroll 4
    for (int kk = 0; kk < KC; kk += 4) {
      const int kn = (kk + 4 < KC) ? kk + 4 : kk;   // last iter reloads
      const float* hpn = Hs + (size_t)(kb + kn + basek) * D;
#pragma unroll
      for (int t = 0; t < 4; ++t) {
        const int col = ncol0 + t * 16 + nlane;
        bn[t].x = hpn[col];
        bn[t].y = hpn[D + col];
      }
      v2f a = *(const v2f*)(lbuf + nlane * LROW + kk + basek);
#pragma unroll
      for (int t = 0; t < 4; ++t) acc[t] = wmma4(a, b[t], acc[t]);
#pragma unroll
      for (int t = 0; t < 4; ++t) b[t] = bn[t];
    }
    __syncthreads();   // all waves done reading this buffer
  }

  const int rowoff = (lane >> 4) * 8;
#pragma unroll
  for (int r = 0; r < 8; ++r) {
    const int row = m0 + rowoff + r;
    const float dv = dinv[row];
#pragma unroll
    for (int t = 0; t < 4; ++t) {
      const int col = ncol0 + t * 16 + nlane;
      float g = dv * acc[t][r] + S[(size_t)row * D + col];
      g = fmaxf(g, 0.f);
      out[(size_t)row * D + col] = X[(size_t)row * D + col] + 0.5f * g;
    }
  }
}

// ---------------------------------------------------------------------------
extern "C" void kernel_launch(void* const* d_in, const int* in_sizes, int n_in,
                              void* d_out, int out_size, void* d_ws, size_t ws_size,
                              hipStream_t stream) {
  (void)in_sizes; (void)n_in; (void)out_size; (void)ws_size;
  const float* A  = (const float*)d_in[0];  // 8192x8192
  const float* X  = (const float*)d_in[1];  // 8192x512
  const float* Wg = (const float*)d_in[2];  // 512x512
  const float* Ws = (const float*)d_in[3];  // 512x512

  // Workspace layout (~25.2 MB): dinv | Hs | S
  float* dinv = (float*)d_ws;
  float* Hs   = dinv + N;
  float* S    = Hs + (size_t)N * D;
  float* out  = (float*)d_out;

  k_degree<<<N, 256, 0, stream>>>(A, dinv);
  // Tiles 0..8191: rows < B (both GEMMs). Tiles 8192..16383: rows >= B.
  k_features<true ><<<1024, 256, 0, stream>>>(X, Wg, Ws, dinv, Hs, S, 0);
  k_features<false><<<1024, 256, 0, stream>>>(X, Wg, Ws, dinv, Hs, S, 8192);
  k_gcn<<<B / 16, 256, 0, stream>>>(A, X, Hs, S, dinv, out);
}